// GenTargets_85014582657516
// MI455X (gfx1250) — compile-verified
//
#include <hip/hip_runtime.h>
#include <stdint.h>

// ---------------------------------------------------------------------------
// FCOS target generation, fully fused across the 5 pyramid levels.
// B=8, M=50 boxes, HW_TOTAL = 12800+3200+800+208+56 = 17064 locations.
// Key simplification: (l+r)*(t+b) == (x2-x1)*(y2-y1)  -> per-box constant area.
// Box data is staged into LDS with gfx1250 async-to-LDS DMA (ASYNCcnt path).
// ---------------------------------------------------------------------------

#define NBOX      50
#define NBATCH    8
#define HW_TOTAL  17064
#define BIG_AREA  99999999.0f

__global__ __launch_bounds__(256) void fcos_targets_kernel(
    const float* __restrict__ gt_boxes,   // [B, 50, 4]
    const int*   __restrict__ classes,    // [B, 50]
    float* __restrict__ out_cls,          // [B, HW]   (int targets as f32)
    float* __restrict__ out_cnt,          // [B, HW]
    float* __restrict__ out_reg,          // [B, HW, 4]
    float* __restrict__ out_coords)       // [HW, 2]
{
    __shared__ __align__(16) float rawBox[NBOX * 4];
    __shared__ int   rawCls[NBOX];
    __shared__ float bx1[NBOX], by1[NBOX], bx2[NBOX], by2[NBOX];
    __shared__ float bcx[NBOX], bcy[NBOX], barea[NBOX];

    const int b = blockIdx.y;
    const int t = threadIdx.x;

    // ---- Stage this batch's boxes + classes into LDS via async DMA ----
    if (t < NBOX) {
        uint32_t lds_box = (uint32_t)(uintptr_t)(&rawBox[t * 4]);
        uint64_t g_box   = (uint64_t)(uintptr_t)gt_boxes
                         + (uint64_t)((b * NBOX + t) * 16);
        asm volatile("global_load_async_to_lds_b128 %0, %1, off"
                     :: "v"(lds_box), "v"(g_box) : "memory");

        uint32_t lds_cls = (uint32_t)(uintptr_t)(&rawCls[t]);
        uint64_t g_cls   = (uint64_t)(uintptr_t)classes
                         + (uint64_t)((b * NBOX + t) * 4);
        asm volatile("global_load_async_to_lds_b32 %0, %1, off"
                     :: "v"(lds_cls), "v"(g_cls) : "memory");
    }
    asm volatile("s_wait_asynccnt 0x0" ::: "memory");
    __syncthreads();

    // ---- Per-box derived quantities (center + area are pixel-invariant) ----
    if (t < NBOX) {
        const float x1 = rawBox[t * 4 + 0], y1 = rawBox[t * 4 + 1];
        const float x2 = rawBox[t * 4 + 2], y2 = rawBox[t * 4 + 3];
        bx1[t] = x1; by1[t] = y1; bx2[t] = x2; by2[t] = y2;
        bcx[t] = 0.5f * (x1 + x2);
        bcy[t] = 0.5f * (y1 + y2);
        barea[t] = (x2 - x1) * (y2 - y1);
    }
    __syncthreads();

    const int p = blockIdx.x * blockDim.x + t;
    if (p >= HW_TOTAL) return;

    // ---- Decode pyramid level; widths are powers of two (128..8) ----
    int lvl, base;
    if      (p < 12800) { lvl = 0; base = 0;     }
    else if (p < 16000) { lvl = 1; base = 12800; }
    else if (p < 16800) { lvl = 2; base = 16000; }
    else if (p < 17008) { lvl = 3; base = 16800; }
    else                { lvl = 4; base = 17008; }

    const int   local = p - base;
    const int   logw  = 7 - lvl;                 // w = 128 >> lvl
    const int   ix    = local & ((1 << logw) - 1);
    const int   iy    = local >> logw;
    const float s     = (float)(8 << lvl);       // stride 8..128
    const float lim0  = (lvl == 0) ? -1.0f     : (float)(32 << lvl);
    const float lim1  = (lvl == 4) ? 999999.0f : (float)(64 << lvl);
    const float radiu = 1.5f * s;
    const float x     = (float)ix * s + 0.5f * s;
    const float y     = (float)iy * s + 0.5f * s;

    // ---- Scan boxes: masked argmin over per-box area ----
    float minA = BIG_AREA;
    int   best = 0;
    bool  any  = false;
    #pragma unroll 5
    for (int m = 0; m < NBOX; ++m) {
        const float l  = x - bx1[m];
        const float tt = y - by1[m];
        const float r  = bx2[m] - x;
        const float bb = by2[m] - y;
        const float omin = fminf(fminf(l, tt), fminf(r, bb));
        const float omax = fmaxf(fmaxf(l, tt), fmaxf(r, bb));
        const float coff = fmaxf(fabsf(x - bcx[m]), fabsf(y - bcy[m]));
        const bool pos = (omin > 0.0f) && (omax > lim0) && (omax <= lim1)
                         && (coff < radiu);
        const float a = pos ? barea[m] : BIG_AREA;
        if (a < minA) { minA = a; best = m; }   // strict '<' == argmin tie rule
        any = any || pos;
    }

    // ---- Emit targets ----
    float  cls_o = 0.0f;
    float  cnt_o = -1.0f;
    float4 reg_o = make_float4(-1.0f, -1.0f, -1.0f, -1.0f);
    if (any) {
        const float l  = x - bx1[best];
        const float tt = y - by1[best];
        const float r  = bx2[best] - x;
        const float bb = by2[best] - y;
        const float l_ = fmaxf(l, 0.0f),  t_ = fmaxf(tt, 0.0f);
        const float r_ = fmaxf(r, 0.0f),  b_ = fmaxf(bb, 0.0f);
        const float lrmin = fminf(l_, r_), lrmax = fmaxf(l_, r_);
        const float tbmin = fminf(t_, b_), tbmax = fmaxf(t_, b_);
        cnt_o = sqrtf(lrmin * tbmin / (lrmax * tbmax + 1e-10f));
        cls_o = (float)rawCls[best];
        reg_o = make_float4(l, tt, r, bb);
    }

    const int o = b * HW_TOTAL + p;
    out_cls[o] = cls_o;
    out_cnt[o] = cnt_o;
    ((float4*)out_reg)[o] = reg_o;               // 16B coalesced store
    if (b == 0) {
        out_coords[2 * p + 0] = x;
        out_coords[2 * p + 1] = y;
    }
}

extern "C" void kernel_launch(void* const* d_in, const int* in_sizes, int n_in,
                              void* d_out, int out_size, void* d_ws, size_t ws_size,
                              hipStream_t stream) {
    (void)in_sizes; (void)n_in; (void)out_size; (void)d_ws; (void)ws_size;
    // setup_inputs order: cls0..cls4 (unused, shape-only), gt_boxes, classes
    const float* gt_boxes = (const float*)d_in[5];
    const int*   classes  = (const int*)d_in[6];

    float* out        = (float*)d_out;
    float* out_cls    = out;                          // 136512
    float* out_cnt    = out + NBATCH * HW_TOTAL;      // 136512
    float* out_reg    = out + 2 * NBATCH * HW_TOTAL;  // 546048
    float* out_coords = out + 6 * NBATCH * HW_TOTAL;  // 34128

    dim3 grid((HW_TOTAL + 255) / 256, NBATCH, 1);
    dim3 block(256, 1, 1);
    fcos_targets_kernel<<<grid, block, 0, stream>>>(
        gt_boxes, classes, out_cls, out_cnt, out_reg, out_coords);
}